// NIDLoss_35510789603822
// MI455X (gfx1250) — compile-verified
//
#include <hip/hip_runtime.h>
#include <hip/hip_bf16.h>

typedef __attribute__((ext_vector_type(2))) float v2f;
typedef __attribute__((ext_vector_type(8))) float v8f;

#define NBATCH 8
#define KB 16
#define CB 4
#define BETA_F 500.0f
#define EPS_SM 1e-12f
#define EPS_F  1e-7f
// half-bandwidth-scaled edge constants:
// camera: edge_k = (g - k/16) / (2*0.005) = g*100 - k*6.25
// label:  edge_c = (la - c + 0.5) / (2*0.001) = la*500 + 250 - c*500
#define CAM_SCL 100.0f
#define CAM_STP 6.25f
#define LAB_SCL 500.0f

// gfx1250 v_tanh_f32: sig(a)-sig(b) == 0.5*(tanh(a/2)-tanh(b/2)) exactly;
// the uniform 0.5 factors on P_c / P_l cancel under the reference's
// re-normalizations, so bins accumulate raw tanh differences.
__device__ __forceinline__ float fast_tanh(float x) {
  float r;
  asm("v_tanh_f32 %0, %1" : "=v"(r) : "v"(x));
  return r;
}

__global__ void nid_init_ws(float* __restrict__ ws) {
  int i = threadIdx.x;
  if (i < 128) ws[i] = 0.0f;
}

// ws layout: [0..63] p_cl(16x4), [64..79] p_c(16), [80..83] p_l(4)
__global__ __launch_bounds__(256) void nid_hist(const float* __restrict__ cam,
                                                const float* __restrict__ lab,
                                                float* __restrict__ ws,
                                                int npix) {
  __shared__ float lds_pc[8][32][KB];  // [wave][local pos][camera bin]
  __shared__ float lds_pl[8][32][CB];  // [wave][local pos][label bin]

  // readfirstlane => wave id is SGPR => scalar loop control + scalar LDS slice
  const int wave   = __builtin_amdgcn_readfirstlane(threadIdx.x >> 5);
  const int lane   = threadIdx.x & 31;
  const int hi     = lane >> 4;   // half-wave select (K pair for WMMA operands)
  const int row    = lane & 15;   // M index (A) / column index (B)
  const int gwave  = blockIdx.x * 8 + wave;
  const int nwaves = gridDim.x * 8;
  const int step   = nwaves * 32;

  v8f acc = {};  // 16x16 f32 C/D: cols 0..3 = p_cl, col 4 = p_c (B col4 == 1.0)
  float plt0 = 0.f, plt1 = 0.f, plt2 = 0.f, plt3 = 0.f;

  for (int base = gwave * 32; base < npix; base += step) {
    const int pos = base + lane;  // per-lane 32-bit element index

    float pc[KB];
#pragma unroll
    for (int k = 0; k < KB; ++k) pc[k] = 0.f;
    float pl[CB];
#pragma unroll
    for (int c = 0; c < CB; ++c) pl[c] = 0.f;

#pragma unroll
    for (int b = 0; b < NBATCH; ++b) {
      // uniform int channel offsets (SALU, loop-invariant) + 32-bit voffset
      const int cb0 = (3 * b + 0) * npix;
      const int cb1 = (3 * b + 1) * npix;
      const int cb2 = (3 * b + 2) * npix;
      float g = (cam[cb0 + pos] + cam[cb1 + pos] + cam[cb2 + pos]) * (1.0f / 3.0f);
      // 16 soft bins from 17 shared tanh edges
      float gh = g * CAM_SCL;
      float t_prev = fast_tanh(gh);
#pragma unroll
      for (int k = 0; k < KB; ++k) {
        float t_next = fast_tanh(gh - CAM_STP * (float)(k + 1));
        pc[k] += t_prev - t_next;
        t_prev = t_next;
      }
      // label soft-argmax over 4 channels, beta = 500
      const int lb0 = (4 * b + 0) * npix;
      const int lb1 = (4 * b + 1) * npix;
      const int lb2 = (4 * b + 2) * npix;
      const int lb3 = (4 * b + 3) * npix;
      float l0 = lab[lb0 + pos], l1 = lab[lb1 + pos];
      float l2 = lab[lb2 + pos], l3 = lab[lb3 + pos];
      float m  = fmaxf(fmaxf(l0, l1), fmaxf(l2, l3));
      float x0 = __expf((l0 - m) * BETA_F);
      float x1 = __expf((l1 - m) * BETA_F);
      float x2 = __expf((l2 - m) * BETA_F);
      float x3 = __expf((l3 - m) * BETA_F);
      float la = (x1 + 2.f * x2 + 3.f * x3) * __frcp_rn(x0 + x1 + x2 + x3 + EPS_SM);
      // 4 soft bins from 5 shared tanh edges
      float lh = la * LAB_SCL + 250.0f;
      float f_prev = fast_tanh(lh);
#pragma unroll
      for (int c = 0; c < CB; ++c) {
        float f_next = fast_tanh(lh - LAB_SCL * (float)(c + 1));
        pl[c] += f_prev - f_next;
        f_prev = f_next;
      }
    }

    // stage per-position histograms for WMMA operand gather
#pragma unroll
    for (int k = 0; k < KB; ++k) lds_pc[wave][lane][k] = pc[k];
#pragma unroll
    for (int c = 0; c < CB; ++c) lds_pl[wave][lane][c] = pl[c];
    plt0 += pl[0]; plt1 += pl[1]; plt2 += pl[2]; plt3 += pl[3];

    // LDS is in-order within a wave; fence compiler + drain DS returns
    asm volatile("s_wait_dscnt 0x0" ::: "memory");

    // 8 WMMAs x (K=4 positions) cover the 32 staged positions
#pragma unroll
    for (int gi = 0; gi < 8; ++gi) {
      const int p0 = gi * 4;
      // A (16x4 f32): lanes 0-15 hold M=row, K=0,1; lanes 16-31 hold K=2,3
      v2f A, B;
      A.x = lds_pc[wave][p0 + 2 * hi + 0][row];
      A.y = lds_pc[wave][p0 + 2 * hi + 1][row];
      // B (4x16 f32): VGPR0 = rows {0|2}, VGPR1 = rows {1|3}; col 4 = 1.0 => p_c
      float b0 = (row < CB) ? lds_pl[wave][p0 + 2 * hi + 0][row] : 0.0f;
      float b1 = (row < CB) ? lds_pl[wave][p0 + 2 * hi + 1][row] : 0.0f;
      B.x = (row == CB) ? 1.0f : b0;
      B.y = (row == CB) ? 1.0f : b1;
      acc = __builtin_amdgcn_wmma_f32_16x16x4_f32(false, A, false, B, (short)0, acc,
                                                  false, false);
    }
    asm volatile("s_wait_dscnt 0x0" ::: "memory");  // WAR guard vs next iter stores
  }

  // wave-reduce p_l partials (wave32)
#pragma unroll
  for (int off = 16; off > 0; off >>= 1) {
    plt0 += __shfl_down(plt0, off, 32);
    plt1 += __shfl_down(plt1, off, 32);
    plt2 += __shfl_down(plt2, off, 32);
    plt3 += __shfl_down(plt3, off, 32);
  }
  float* ws_pcl = ws;
  float* ws_pc  = ws + 64;
  float* ws_pl  = ws + 80;
  if (lane == 0) {
    atomicAdd(&ws_pl[0], plt0);
    atomicAdd(&ws_pl[1], plt1);
    atomicAdd(&ws_pl[2], plt2);
    atomicAdd(&ws_pl[3], plt3);
  }
  // D layout: VGPR v holds rows v (lanes 0-15) and v+8 (lanes 16-31), col = lane&15
  if (row <= CB) {
#pragma unroll
    for (int v = 0; v < 8; ++v) {
      const int m  = v + 8 * hi;
      float val = acc[v];
      if (row < CB) atomicAdd(&ws_pcl[m * CB + row], val);
      else          atomicAdd(&ws_pc[m], val);
    }
  }
}

__global__ void nid_final(const float* __restrict__ ws, float* __restrict__ out) {
  if (threadIdx.x != 0 || blockIdx.x != 0) return;
  float pcl[KB][CB], pcv[KB], plv[CB];
  float scl = 0.f, sc = 0.f, sl = 0.f;
  for (int k = 0; k < KB; ++k)
    for (int c = 0; c < CB; ++c) { pcl[k][c] = ws[k * CB + c]; scl += pcl[k][c]; }
  for (int k = 0; k < KB; ++k) { pcv[k] = ws[64 + k]; sc += pcv[k]; }
  for (int c = 0; c < CB; ++c) { plv[c] = ws[80 + c]; sl += plv[c]; }
  // the reference's /norm divisions (and the dropped 0.5 tanh factors)
  // cancel under re-normalization
  float iscl = __frcp_rn(scl), isc = __frcp_rn(sc), isl = __frcp_rn(sl);
  float I = 0.f, H = 0.f;
  for (int k = 0; k < KB; ++k) {
    for (int c = 0; c < CB; ++c) {
      float q  = pcl[k][c] * iscl;
      float lq = __logf(q + EPS_F);
      float o  = (pcv[k] * isc) * (plv[c] * isl);
      I += q * (lq - __logf(o + EPS_F));
      H -= q * lq;
    }
  }
  float nid = 1.0f - I * __frcp_rn(H);
  out[0] = (nid - 0.95f) * 20.0f;
}

extern "C" void kernel_launch(void* const* d_in, const int* in_sizes, int n_in,
                              void* d_out, int out_size, void* d_ws, size_t ws_size,
                              hipStream_t stream) {
  const float* cam = (const float*)d_in[0];  // (8,3,512,1024) fp32
  const float* lab = (const float*)d_in[1];  // (8,4,512,1024) fp32
  float* ws  = (float*)d_ws;
  float* out = (float*)d_out;
  const int npix = in_sizes[0] / (NBATCH * 3);  // 512*1024

  nid_init_ws<<<1, 128, 0, stream>>>(ws);
  // 512 blocks x 8 waves = 4096 waves; 524288 / (4096*32) = 4 uniform chunks/wave
  nid_hist<<<512, 256, 0, stream>>>(cam, lab, ws, npix);
  nid_final<<<1, 32, 0, stream>>>(ws, out);
}